// RelGraphNN_88399016886801
// MI455X (gfx1250) — compile-verified
//
#include <hip/hip_runtime.h>
#include <hip/hip_bf16.h>

typedef __attribute__((ext_vector_type(2))) float v2f;
typedef __attribute__((ext_vector_type(4))) float v4f;
typedef __attribute__((ext_vector_type(8))) float v8f;

#define HID 128          // hidden width (all layer outputs)
#define EPSBN 1e-5f
#define KCH 64           // K-chunk staged in LDS (K is 64 or 128)
#define BSTR 144         // padded LDS row stride in dwords: row shift = 32 banks
                         // -> lanes 0-15 (row ka) and 16-31 (row ka+2) hit
                         //    disjoint bank ranges within one ds_load

// ---------------------------------------------------------------------------
// GEMM: C[z] = A @ B[z] (+ bias), A:[M,K] row-major, B:[K,128] row-major.
// B tile staged in LDS per block; one wave computes a 16x128 strip using
// 8 x v_wmma_f32_16x16x4_f32 per K-step. Block = 256 threads = 8 waves.
// blockIdx.z selects weight slice (relation) via sliceB/sliceC strides.
// ---------------------------------------------------------------------------
__global__ __launch_bounds__(256)
void gemm_wmma_kernel(const float* __restrict__ A,
                      const float* __restrict__ Bw,
                      const float* __restrict__ bias,   // may be nullptr
                      float* __restrict__ C,
                      int M, int K,
                      long long sliceB, long long sliceC)
{
    __shared__ float Bs[KCH * BSTR];                    // 36,864 bytes

    const int z    = blockIdx.z;
    const float* B = Bw + (long long)z * sliceB;
    float* Cz      = C  + (long long)z * sliceC;

    const int lane = threadIdx.x & 31;
    const int wave = threadIdx.x >> 5;
    const int col  = lane & 15;       // output column within 16-wide subtile
    const int hi   = lane >> 4;       // upper half-wave selector

    const int rowBase = blockIdx.x * 128 + wave * 16;
    int rowA = rowBase + col;         // A-fragment row (M = lane%16)
    if (rowA >= M) rowA = M - 1;      // clamp loads; stores are guarded

    v8f acc[8];
#pragma unroll
    for (int n = 0; n < 8; ++n) {
        float bv = bias ? bias[n * 16 + col] : 0.0f;
#pragma unroll
        for (int j = 0; j < 8; ++j) acc[n][j] = bv;
    }

    // LDS copy mapping: thread t covers row t/4 (0..63), 32-col quarter (t%4)
    const int crow = threadIdx.x >> 2;
    const int ccol = (threadIdx.x & 3) * 32;

    const float* Arow = A + (long long)rowA * K;

    for (int kb = 0; kb < K; kb += KCH) {
        // ---- stage B[kb .. kb+63][0..127] into LDS (coalesced 16B) ----
        {
            const float* gsrc = B + (long long)(kb + crow) * HID + ccol;
            float* ldst = &Bs[crow * BSTR + ccol];
#pragma unroll
            for (int i = 0; i < 8; ++i) {
                v4f v = *(const v4f*)(gsrc + 4 * i);
                *(v4f*)(ldst + 4 * i) = v;
            }
        }
        __syncthreads();

        // ---- compute on the chunk ----
#pragma unroll 4
        for (int k0 = 0; k0 < KCH; k0 += 4) {
            const int ka = k0 + 2 * hi;   // lanes 0-15: K=k0,k0+1 ; 16-31: k0+2,k0+3
            v2f a;
            a.x = Arow[kb + ka];
            a.y = Arow[kb + ka + 1];
            const float* Bk0 = &Bs[ka * BSTR];
            const float* Bk1 = Bk0 + BSTR;
#pragma unroll
            for (int n = 0; n < 8; ++n) {
                v2f b;
                b.x = Bk0[n * 16 + col];
                b.y = Bk1[n * 16 + col];
                acc[n] = __builtin_amdgcn_wmma_f32_16x16x4_f32(
                    false, a, false, b, (short)0, acc[n], false, false);
            }
        }
        __syncthreads();
    }

#pragma unroll
    for (int n = 0; n < 8; ++n) {
#pragma unroll
        for (int v = 0; v < 8; ++v) {
            int row = rowBase + v + 8 * hi;   // C layout: VGPR v -> M=v (+8 for hi)
            if (row < M)
                Cz[(long long)row * HID + n * 16 + col] = acc[n][v];
        }
    }
}

// ---------------------------------------------------------------------------
// Edge scatter: 32 threads per edge; non-temporal 16B gather from
// h_all[etype][src] (single-use stream, keep out of L2), 4 coalesced
// global f32 atomics into agg[dst]. agg pre-initialized to h@loopW + bias.
// ---------------------------------------------------------------------------
__global__ __launch_bounds__(256)
void scatter_kernel(const float* __restrict__ hall,
                    const int* __restrict__ src,
                    const int* __restrict__ dst,
                    const int* __restrict__ et,
                    float* __restrict__ agg,
                    int N, long long E)
{
    long long g = (long long)blockIdx.x * blockDim.x + threadIdx.x;
    if (g >= E * 32) return;
    long long e = g >> 5;
    int c = (int)(g & 31);                      // 32 chunks of 4 floats = 128
    int s = src[e], d = dst[e], r = et[e];
    const v4f* p = (const v4f*)(hall + (((long long)r * N + s) << 7) + 4 * c);
    const v4f v = __builtin_nontemporal_load(p);
    float* out = agg + ((long long)d << 7) + 4 * c;
    atomicAdd(out + 0, v.x);
    atomicAdd(out + 1, v.y);
    atomicAdd(out + 2, v.z);
    atomicAdd(out + 3, v.w);
}

__global__ void zero_stats_kernel(float* stats)
{
    stats[threadIdx.x] = 0.0f;   // 256 floats: [sum(128) | sumsq(128)]
}

// Per-feature sum / sum-of-squares over the node dimension.
__global__ __launch_bounds__(128)
void bn_stats_kernel(const float* __restrict__ h, float* __restrict__ stats, int N)
{
    int f = threadIdx.x;                        // feature 0..127
    float s1 = 0.0f, s2 = 0.0f;
    for (int r = blockIdx.x; r < N; r += gridDim.x) {
        float v = h[(long long)r * HID + f];
        s1 += v;
        s2 += v * v;
    }
    atomicAdd(&stats[f], s1);
    atomicAdd(&stats[HID + f], s2);
}

__global__ __launch_bounds__(256)
void bn_norm_kernel(const float* __restrict__ h, const float* __restrict__ stats,
                    const float* __restrict__ gamma, const float* __restrict__ beta,
                    float* __restrict__ out, int N, int relu)
{
    long long i = (long long)blockIdx.x * blockDim.x + threadIdx.x;
    if (i >= (long long)N * HID) return;
    int f = (int)(i & (HID - 1));
    float invN = 1.0f / (float)N;
    float mean = stats[f] * invN;
    float var  = stats[HID + f] * invN - mean * mean;
    float v = gamma[f] * (h[i] - mean) * rsqrtf(var + EPSBN) + beta[f];
    if (relu) v = fmaxf(v, 0.0f);
    out[i] = v;
}

// ---------------------------------------------------------------------------
// Host orchestration
// ---------------------------------------------------------------------------
static inline void run_bn(const float* aggp, float* stats,
                          const float* g, const float* be,
                          float* outp, int N, int relu, hipStream_t stream)
{
    zero_stats_kernel<<<1, 256, 0, stream>>>(stats);
    bn_stats_kernel<<<512, 128, 0, stream>>>(aggp, stats, N);
    long long tot = (long long)N * HID;
    bn_norm_kernel<<<(int)((tot + 255) / 256), 256, 0, stream>>>(
        aggp, stats, g, be, outp, N, relu);
}

extern "C" void kernel_launch(void* const* d_in, const int* in_sizes, int n_in,
                              void* d_out, int out_size, void* d_ws, size_t ws_size,
                              hipStream_t stream)
{
    const int F_in = 64;
    const int N = in_sizes[0] / F_in;
    const long long E = in_sizes[1];
    const int R = in_sizes[4] / (F_in * HID);          // 4
    const int L = in_sizes[9] / (R * HID * HID);       // 2

    const float* x     = (const float*)d_in[0];
    const int*   src   = (const int*)d_in[1];
    const int*   dst   = (const int*)d_in[2];
    const int*   ety   = (const int*)d_in[3];
    const float* W0    = (const float*)d_in[4];
    const float* loop0 = (const float*)d_in[5];
    const float* b0    = (const float*)d_in[6];
    const float* g0    = (const float*)d_in[7];
    const float* be0   = (const float*)d_in[8];
    const float* Ws    = (const float*)d_in[9];
    const float* loops = (const float*)d_in[10];
    const float* bs    = (const float*)d_in[11];
    const float* gs    = (const float*)d_in[12];
    const float* bes   = (const float*)d_in[13];
    const float* Wl    = (const float*)d_in[14];
    const float* bl    = (const float*)d_in[15];
    const float* gl    = (const float*)d_in[16];
    const float* bel   = (const float*)d_in[17];

    // Workspace layout (floats)
    float* w     = (float*)d_ws;
    long long NH = (long long)N * HID;
    float* hall  = w;                    // R * N * H
    float* agg   = hall + (long long)R * NH;
    float* hA    = agg + NH;
    float* hB    = hA + NH;
    float* stats = hB + NH;              // 256 floats

    const int Mblocks = (N + 127) / 128;
    const long long sT = E * 32;
    const int sBlocks = (int)((sT + 255) / 256);

    // ---- Layer 0: x[64] -> hA[128] ----
    {
        dim3 gR(Mblocks, 1, R);
        gemm_wmma_kernel<<<gR, 256, 0, stream>>>(x, W0, nullptr, hall,
                                                 N, F_in,
                                                 (long long)F_in * HID, NH);
        gemm_wmma_kernel<<<dim3(Mblocks, 1, 1), 256, 0, stream>>>(
            x, loop0, b0, agg, N, F_in, 0, 0);
        scatter_kernel<<<sBlocks, 256, 0, stream>>>(hall, src, dst, ety, agg, N, E);
        run_bn(agg, stats, g0, be0, hA, N, 1, stream);
    }

    // ---- Hidden layers: 128 -> 128 ----
    const float* hin = hA;
    float* hout = hB;
    for (int l = 0; l < L; ++l) {
        const float* Wl_r  = Ws    + (long long)l * R * HID * HID;
        const float* loopl = loops + (long long)l * HID * HID;
        const float* bsl   = bs  + (long long)l * HID;
        const float* gsl   = gs  + (long long)l * HID;
        const float* besl  = bes + (long long)l * HID;

        dim3 gR(Mblocks, 1, R);
        gemm_wmma_kernel<<<gR, 256, 0, stream>>>(hin, Wl_r, nullptr, hall,
                                                 N, HID,
                                                 (long long)HID * HID, NH);
        gemm_wmma_kernel<<<dim3(Mblocks, 1, 1), 256, 0, stream>>>(
            hin, loopl, bsl, agg, N, HID, 0, 0);
        scatter_kernel<<<sBlocks, 256, 0, stream>>>(hall, src, dst, ety, agg, N, E);
        run_bn(agg, stats, gsl, besl, hout, N, 1, stream);

        const float* t = hin; hin = hout; hout = (float*)t;
    }

    // ---- Final: Linear + BN (no ReLU) -> d_out ----
    gemm_wmma_kernel<<<dim3(Mblocks, 1, 1), 256, 0, stream>>>(
        hin, Wl, bl, agg, N, HID, 0, 0);
    run_bn(agg, stats, gl, bel, (float*)d_out, N, 0, stream);
}